// UnfoldNd_14267881357345
// MI455X (gfx1250) — compile-verified
//
#include <hip/hip_runtime.h>

// UnfoldNd (im2col, K=3, stride=1, pad=1, dil=1) for x[16,64,128,128] f32
// -> out[16, 576, 16384] f32.  Pure data movement: 64 MiB read, 576 MiB write.
// Roofline: ~640 MiB @ 23.3 TB/s => ~27.5 us; WMMA irrelevant (one-hot GEMM is a copy).
// CDNA5 path: async global->LDS staging (ASYNCcnt) + non-temporal streaming stores.

typedef float v4f __attribute__((ext_vector_type(4)));
typedef int   v4i __attribute__((vector_size(16)));   // matches builtin prototype

constexpr int Bn = 16, Cn = 64, H = 128, W = 128;
constexpr int Pn = 128, Qn = 128;
constexpr int TP = 8;              // output rows (p) per block
constexpr int LROWS = TP + 2;      // input rows staged in LDS (with row halo)
constexpr int LSTR  = 136;         // LDS row stride (floats); body at +4 is 16B-aligned
constexpr int NPLANES = Bn * Cn;   // 1024
constexpr int NTILES  = Pn / TP;   // 16

__global__ __launch_bounds__(256) void unfold3x3_kernel(const float* __restrict__ x,
                                                        float* __restrict__ out) {
    __shared__ float lds[LROWS * LSTR];   // 10*136*4 = 5440 bytes

    const int tid  = threadIdx.x;
    const int wave = tid >> 5;            // 0..7 (wave32)
    const int lane = tid & 31;

    const int plane = (int)blockIdx.x >> 4;     // b*C + c
    const int tile  = (int)blockIdx.x & 15;
    const int p0    = tile * TP;

    // ---- 1) zero-fill LDS: halo columns + any out-of-range rows become the
    //         conv zero-padding for free -------------------------------------
    for (int i = tid; i < LROWS * LSTR; i += 256) lds[i] = 0.0f;
    __syncthreads();

    // ---- 2) stage valid input rows into LDS body (async global -> LDS) -----
    const float* xplane = x + (size_t)plane * (H * W);

#if __has_builtin(__builtin_amdgcn_global_load_async_to_lds_b128)
    for (int j = wave; j < LROWS; j += 8) {
        const int r = p0 - 1 + j;
        if (r >= 0 && r < H) {
            __attribute__((address_space(1))) v4i* gsrc =
                (__attribute__((address_space(1))) v4i*)(xplane + (size_t)r * W + 4 * lane);
            __attribute__((address_space(3))) v4i* ldst =
                (__attribute__((address_space(3))) v4i*)(&lds[j * LSTR + 4 + 4 * lane]);
            // per-lane 16B: one wave instruction moves a full 512B row
            __builtin_amdgcn_global_load_async_to_lds_b128(gsrc, ldst, 0, 0);
        }
    }
  #if __has_builtin(__builtin_amdgcn_s_wait_asynccnt)
    __builtin_amdgcn_s_wait_asynccnt(0);
  #else
    asm volatile("s_wait_asynccnt 0" ::: "memory");
  #endif
#else
    // synchronous fallback: global_load_b128 + ds_store_b128
    for (int j = wave; j < LROWS; j += 8) {
        const int r = p0 - 1 + j;
        if (r >= 0 && r < H) {
            v4f v = *(const v4f*)(xplane + (size_t)r * W + 4 * lane);
            *(v4f*)(&lds[j * LSTR + 4 + 4 * lane]) = v;
        }
    }
#endif
    __syncthreads();

    // ---- 3) emit 9 shifted copies; one wave per 512B output row segment ----
    // 72 row-instances per block = 8 p-rows x 9 (k,l); wave w takes idx = w+8*it
    float* obase = out + (size_t)plane * (9 * Pn * Qn);

#pragma unroll
    for (int it = 0; it < 9; ++it) {
        const int idx   = wave + it * 8;      // 0..71
        const int p_off = idx / 9;
        const int kl    = idx - 9 * p_off;    // k*3 + l
        const int k     = kl / 3;
        const int l     = kl - 3 * k;
        const int j     = p_off + k;          // LDS row holding input row p+k-1

        // source cols (q+l-1) for q = 4*lane .. 4*lane+3  (shift -> scalar ds reads)
        const float* s = &lds[j * LSTR + 3 + l + 4 * lane];
        v4f v = { s[0], s[1], s[2], s[3] };

        float* o = obase + (size_t)kl * (Pn * Qn) + (size_t)(p0 + p_off) * Qn + 4 * lane;
        __builtin_nontemporal_store(v, (v4f*)o);   // 576 MiB write-once: stream past L2
    }
}

extern "C" void kernel_launch(void* const* d_in, const int* in_sizes, int n_in,
                              void* d_out, int out_size, void* d_ws, size_t ws_size,
                              hipStream_t stream) {
    (void)in_sizes; (void)n_in; (void)out_size; (void)d_ws; (void)ws_size;
    const float* x = (const float*)d_in[0];
    float* out = (float*)d_out;
    unfold3x3_kernel<<<dim3(NPLANES * NTILES), dim3(256), 0, stream>>>(x, out);
}